// Decoder_29764123361311
// MI455X (gfx1250) — compile-verified
//
#include <hip/hip_runtime.h>
#include <math.h>

#define NB 8            // batch
#define NA 76725        // anchors
#define NC 80           // classes
#define KC 256          // per-class candidate pool (K_CAND)
#define MAXT 200        // max total detections
#define HBITS 11
#define HSIZE (1u << HBITS)      // 2048 bins
#define HSHIFT (32 - HBITS)      // 21
#define CANDMAX 1024
#define EQBASE 256               // slots [0,EQBASE) reserved for strictly-greater keys
#define TILE2 4096               // anchors per TDM chunk (16 KB LDS, double-buffered)

#if defined(__has_builtin)
#if __has_builtin(__builtin_amdgcn_tensor_load_to_lds) && __has_builtin(__builtin_amdgcn_s_wait_tensorcnt)
#define HAVE_TDM 1
#endif
#endif

typedef unsigned v4u __attribute__((ext_vector_type(4)));
typedef int      v8i __attribute__((ext_vector_type(8)));
typedef int      v4i __attribute__((ext_vector_type(4)));

#define RFL(x) ((unsigned)__builtin_amdgcn_readfirstlane((int)(x)))

// ---------- monotonic float <-> uint key ----------
__device__ __forceinline__ unsigned fkey(float f) {
    unsigned u = __float_as_uint(f);
    return u ^ ((u & 0x80000000u) ? 0xFFFFFFFFu : 0x80000000u);
}
__device__ __forceinline__ float unfkey(unsigned k) {
    unsigned u = (k & 0x80000000u) ? (k ^ 0x80000000u) : ~k;
    return __uint_as_float(u);
}

#ifdef HAVE_TDM
// Issue one TDM descriptor: gather `rows` floats with pitch NC into contiguous LDS.
__device__ __forceinline__ void tdm_issue(const float* gcol, unsigned lds, int rows) {
    unsigned long long ga = (unsigned long long)(uintptr_t)gcol;
    // D# group 0: count=1 | lds_addr | global_addr | type=2
    v4u g0 = { RFL(1u),
               RFL(lds),
               RFL((unsigned)(ga & 0xFFFFFFFFu)),
               RFL(((unsigned)((ga >> 32) & 0x01FFFFFFu)) | (2u << 30)) };
    // D# group 1
    unsigned r = (unsigned)rows;
    v8i g1 = { (int)RFL(2u << 16),                          // wg_mask=0, data_size=2 (4B)
               (int)RFL(1u << 16),                          // atomic_barrier=0 | tensor_dim0.lo16=1
               (int)RFL(r << 16),                           // tensor_dim0.hi16=0 | tensor_dim1.lo16
               (int)RFL(((r >> 16) & 0xFFFFu) | (1u << 16)),// tensor_dim1.hi16 | tile_dim0=1
               (int)RFL(r & 0xFFFFu),                       // tile_dim1=rows, tile_dim2=0
               (int)RFL((unsigned)NC),                      // tensor_dim0_stride = 80 elements
               0, 0 };                                      // dim1_stride unused (2D tile)
    v4i z4 = { 0, 0, 0, 0 };                                // groups 2/3: NULL (2-D tensor)
    v8i z8 = { 0, 0, 0, 0, 0, 0, 0, 0 };
    __builtin_amdgcn_tensor_load_to_lds(g0, g1, z4, z4, z8, 0);
}
#endif

__device__ __forceinline__ int chunk_rows(int ci) {
    int base = ci * TILE2;
    return (NA - base < TILE2) ? (NA - base) : TILE2;
}

// ---------- double-buffered TDM streaming of a strided class column ----------
// proc(tileptr, base, rows) consumes one staged chunk (called under barriers).
template <typename F>
__device__ void stream_col_chunks(const float* col, float (*tiles)[TILE2],
                                  int tid, int nthr, F&& proc) {
    const int nchunks = (NA + TILE2 - 1) / TILE2;
#ifdef HAVE_TDM
    if (tid < 32) tdm_issue(col, (unsigned)(uintptr_t)tiles[0], chunk_rows(0));
    for (int ci = 0; ci < nchunks; ++ci) {
        int cur = ci & 1;
        if (tid < 32) {  // one wave drives the DMA pipeline
            if (ci + 1 < nchunks) {
                tdm_issue(col + (size_t)(ci + 1) * TILE2 * NC,
                          (unsigned)(uintptr_t)tiles[cur ^ 1], chunk_rows(ci + 1));
                __builtin_amdgcn_s_wait_tensorcnt(1);   // chunk ci complete (in-order)
            } else {
                __builtin_amdgcn_s_wait_tensorcnt(0);   // drain
            }
        }
        __syncthreads();                 // chunk ci visible to all waves
        proc(tiles[cur], ci * TILE2, chunk_rows(ci));
        __syncthreads();                 // done reading before buffer reuse
    }
#else
    for (int ci = 0; ci < nchunks; ++ci) {
        int base = ci * TILE2, rows = chunk_rows(ci);
        __syncthreads();
        for (int i = tid; i < rows; i += nthr) tiles[0][i] = col[(size_t)(base + i) * NC];
        __syncthreads();
        proc(tiles[0], base, rows);
    }
    __syncthreads();
#endif
}

// ---------- pad + bitonic sort (descending key, ascending index on ties) ----------
__device__ void pad_and_sort(unsigned* ckey, int* cidx, unsigned g, unsigned e,
                             int tid, int nthr) {
    if (g > EQBASE) g = EQBASE;
    if (e > CANDMAX - EQBASE) e = CANDMAX - EQBASE;
    for (unsigned i = g + tid; i < EQBASE; i += nthr)           { ckey[i] = 0u; cidx[i] = 0x7FFFFFFF; }
    for (unsigned i = EQBASE + e + tid; i < CANDMAX; i += nthr) { ckey[i] = 0u; cidx[i] = 0x7FFFFFFF; }
    for (unsigned kk = 2; kk <= CANDMAX; kk <<= 1) {
        for (unsigned jj = kk >> 1; jj > 0; jj >>= 1) {
            __syncthreads();
            for (unsigned i = tid; i < CANDMAX; i += nthr) {
                unsigned ixj = i ^ jj;
                if (ixj > i) {
                    unsigned ka = ckey[i];   int ia = cidx[i];
                    unsigned kb = ckey[ixj]; int ib = cidx[ixj];
                    bool iAfter = (ka < kb) || (ka == kb && ia > ib);
                    bool descRegion = ((i & kk) == 0);
                    if (descRegion ? iAfter : !iAfter) {
                        ckey[i] = kb;   cidx[i] = ib;
                        ckey[ixj] = ka; cidx[ixj] = ia;
                    }
                }
            }
        }
    }
    __syncthreads();
}

// ---------- kernel 1: box decode ----------
__global__ void decode_kernel(const float* __restrict__ box_pred,
                              const float* __restrict__ anchors,
                              float* __restrict__ dec) {
    int i = blockIdx.x * blockDim.x + threadIdx.x;
    if (i >= NB * NA) return;
    __builtin_prefetch(box_pred + (size_t)i * 4 + 2048, 0, 3);  // near-scope global_prefetch_b8
    int a = i % NA;
    float4 d  = ((const float4*)box_pred)[i];
    float4 an = ((const float4*)anchors)[a];
    float cx = d.x * 0.1f * an.z + an.x;
    float cy = d.y * 0.1f * an.w + an.y;
    float w  = expf(d.z * 0.2f) * an.z;
    float h  = expf(d.w * 0.2f) * an.w;
    float4 o;
    o.x = cx - 0.5f * w; o.y = cy - 0.5f * h;
    o.z = cx + 0.5f * w; o.w = cy + 0.5f * h;
    ((float4*)dec)[i] = o;
}

// ---------- kernel 2: per-(b,c) top-256 via double-buffered TDM LDS tiles ----------
__global__ void __launch_bounds__(512) topk_kernel(const float* __restrict__ cls_pred,
                                                   float* __restrict__ top_sc,
                                                   int* __restrict__ top_ix) {
    __shared__ float    tiles[2][TILE2];
    __shared__ unsigned hist[HSIZE];
    __shared__ unsigned scal[8];
    __shared__ unsigned ckey[CANDMAX];
    __shared__ int      cidx[CANDMAX];
    int bc = blockIdx.x;
    int b = bc / NC, c = bc % NC;
    int tid = threadIdx.x, nthr = blockDim.x;
    const float* col = cls_pred + (size_t)b * NA * NC + c;

    for (int i = tid; i < (int)HSIZE; i += nthr) hist[i] = 0u;
    __syncthreads();

    // pass 1: histogram of top HBITS key bits
    stream_col_chunks(col, tiles, tid, nthr,
        [&](const float* t, int base, int rows) {
            (void)base;
            for (int i = tid; i < rows; i += nthr)
                atomicAdd(&hist[fkey(t[i]) >> HSHIFT], 1u);
        });
    __syncthreads();
    if (tid == 0) {
        unsigned cum = 0, bin = 0;
        for (int i = (int)HSIZE - 1; i >= 0; --i) {
            if (cum + hist[i] >= (unsigned)KC) { bin = (unsigned)i; break; }
            cum += hist[i];
        }
        scal[0] = bin; scal[4] = 0u; scal[5] = 0u;
    }
    __syncthreads();
    unsigned bin = scal[0];

    // pass 2: gather candidates (strictly-greater guaranteed retained)
    stream_col_chunks(col, tiles, tid, nthr,
        [&](const float* t, int base, int rows) {
            for (int i = tid; i < rows; i += nthr) {
                unsigned k = fkey(t[i]);
                unsigned hb = k >> HSHIFT;
                if (hb > bin) {
                    unsigned p = atomicAdd(&scal[4], 1u);
                    if (p < EQBASE) { ckey[p] = k; cidx[p] = base + i; }
                } else if (hb == bin) {
                    unsigned p = atomicAdd(&scal[5], 1u);
                    if (p < CANDMAX - EQBASE) { ckey[EQBASE + p] = k; cidx[EQBASE + p] = base + i; }
                }
            }
        });
    __syncthreads();
    pad_and_sort(ckey, cidx, scal[4], scal[5], tid, nthr);

    for (int t = tid; t < KC; t += nthr) {
        top_sc[(size_t)bc * KC + t] = unfkey(ckey[t]);   // logit
        top_ix[(size_t)bc * KC + t] = cidx[t];
    }
}

// ---------- kernel 3: greedy NMS per (b,c), 256 threads == 256 candidates ----------
__global__ void nms_kernel(const float* __restrict__ dec,
                           const float* __restrict__ top_sc,
                           const int* __restrict__ top_ix,
                           float* __restrict__ sel_sc,
                           float* __restrict__ cand_boxes) {
    __shared__ float4 sbox[KC];
    __shared__ int    keep[KC];
    __shared__ int    cum[KC];
    int bc = blockIdx.x;
    int b = bc / NC, c = bc % NC;
    int j = threadIdx.x;

    float s   = top_sc[(size_t)bc * KC + j];
    int   idx = top_ix[(size_t)bc * KC + j];
    if (idx < 0 || idx >= NA) idx = 0;  // padding guard
    float4 box = ((const float4*)dec)[(size_t)b * NA + idx];
    sbox[j] = box;
    float score = 1.0f / (1.0f + expf(-s));          // sigmoid
    keep[j] = (score > 0.5f) ? 1 : 0;                // SCORE_TH eligibility

    float aj = (box.z - box.x) * (box.w - box.y);
    for (int i = 0; i < KC; ++i) {
        __syncthreads();
        int ki = keep[i];
        if (ki) {
            float4 bi = sbox[i];
            float ltx = fmaxf(bi.x, box.x), lty = fmaxf(bi.y, box.y);
            float rbx = fminf(bi.z, box.z), rby = fminf(bi.w, box.w);
            float iw = fmaxf(rbx - ltx, 0.0f), ih = fmaxf(rby - lty, 0.0f);
            float inter = iw * ih;
            float ai = (bi.z - bi.x) * (bi.w - bi.y);
            float iou = inter / fmaxf(ai + aj - inter, 1e-8f);
            if (j > i && iou > 0.5f) keep[j] = 0;
        }
    }
    __syncthreads();
    if (j == 0) {                    // exclusive prefix of keep -> cap at MAXT
        int acc = 0;
        for (int i = 0; i < KC; ++i) { cum[i] = acc; acc += keep[i]; }
    }
    __syncthreads();
    int ok = keep[j] && (cum[j] < MAXT);
    size_t base = (size_t)b * (NC * KC) + (size_t)c * KC + j;
    sel_sc[base] = ok ? score : -1.0f;
    ((float4*)cand_boxes)[base] = box;
}

// ---------- kernel 4: per-batch top-200 across classes + output ----------
__global__ void final_kernel(const float* __restrict__ sel_sc,
                             const float* __restrict__ cand_boxes,
                             float* __restrict__ out) {
    __shared__ unsigned hist[HSIZE];
    __shared__ unsigned scal[8];
    __shared__ unsigned ckey[CANDMAX];
    __shared__ int      cidx[CANDMAX];
    int b = blockIdx.x;
    int tid = threadIdx.x, nthr = blockDim.x;
    const int N = NC * KC;
    const float* p = sel_sc + (size_t)b * N;

    for (int i = tid; i < (int)HSIZE; i += nthr) hist[i] = 0u;
    __syncthreads();
    for (int a = tid; a < N; a += nthr) atomicAdd(&hist[fkey(p[a]) >> HSHIFT], 1u);
    __syncthreads();
    if (tid == 0) {
        unsigned cum = 0, bin = 0;
        for (int i = (int)HSIZE - 1; i >= 0; --i) {
            if (cum + hist[i] >= (unsigned)MAXT) { bin = (unsigned)i; break; }
            cum += hist[i];
        }
        scal[0] = bin; scal[4] = 0u; scal[5] = 0u;
    }
    __syncthreads();
    unsigned bin = scal[0];
    for (int a = tid; a < N; a += nthr) {
        unsigned k = fkey(p[a]);
        unsigned hb = k >> HSHIFT;
        if (hb > bin) {
            unsigned q = atomicAdd(&scal[4], 1u);
            if (q < EQBASE) { ckey[q] = k; cidx[q] = a; }
        } else if (hb == bin) {
            unsigned q = atomicAdd(&scal[5], 1u);
            if (q < CANDMAX - EQBASE) { ckey[EQBASE + q] = k; cidx[EQBASE + q] = a; }
        }
    }
    __syncthreads();
    pad_and_sort(ckey, cidx, scal[4], scal[5], tid, nthr);

    float* ob = out;                               // boxes  [NB,MAXT,4]
    float* os = out + (size_t)NB * MAXT * 4;       // scores [NB,MAXT]
    float* ol = os + (size_t)NB * MAXT;            // labels [NB,MAXT]
    for (int t = tid; t < MAXT; t += nthr) {
        float fs = unfkey(ckey[t]);
        int   fi = cidx[t];
        bool valid = (fs > 0.0f) && (fi >= 0) && (fi < N);
        float4 bx = make_float4(0.0f, 0.0f, 0.0f, 0.0f);
        float lbl = -1.0f, sc = 0.0f;
        if (valid) {
            bx  = ((const float4*)cand_boxes)[(size_t)b * N + fi];
            lbl = (float)(fi / KC);
            sc  = fs;
        }
        ((float4*)ob)[(size_t)b * MAXT + t] = bx;
        os[(size_t)b * MAXT + t] = sc;
        ol[(size_t)b * MAXT + t] = lbl;
    }
}

extern "C" void kernel_launch(void* const* d_in, const int* in_sizes, int n_in,
                              void* d_out, int out_size, void* d_ws, size_t ws_size,
                              hipStream_t stream) {
    (void)in_sizes; (void)n_in; (void)out_size; (void)ws_size;
    const float* box_pred = (const float*)d_in[0];
    const float* cls_pred = (const float*)d_in[1];
    const float* anchors  = (const float*)d_in[2];
    float* out = (float*)d_out;

    char* ws = (char*)d_ws;
    size_t o = 0;
    float* dec    = (float*)(ws + o); o += (size_t)NB * NA * 4 * sizeof(float);       // decoded corners
    float* top_sc = (float*)(ws + o); o += (size_t)NB * NC * KC * sizeof(float);      // top logits
    int*   top_ix = (int*)  (ws + o); o += (size_t)NB * NC * KC * sizeof(int);        // top indices
    float* sel_sc = (float*)(ws + o); o += (size_t)NB * NC * KC * sizeof(float);      // post-NMS scores
    float* cand   = (float*)(ws + o); o += (size_t)NB * NC * KC * 4 * sizeof(float);  // candidate boxes

    decode_kernel<<<(NB * NA + 255) / 256, 256, 0, stream>>>(box_pred, anchors, dec);
    topk_kernel<<<NB * NC, 512, 0, stream>>>(cls_pred, top_sc, top_ix);
    nms_kernel<<<NB * NC, KC, 0, stream>>>(dec, top_sc, top_ix, sel_sc, cand);
    final_kernel<<<NB, 256, 0, stream>>>(sel_sc, cand, out);
}